// SampleAndGather_54219667144938
// MI455X (gfx1250) — compile-verified
//
#include <hip/hip_runtime.h>
#include <cstdint>

// Problem shape (fixed by the reference's setup_inputs)
constexpr int kB = 8;
constexpr int kN = 32768;
constexpr int kS = 1024;   // NUM_POINT
constexpr int kC = 96;

constexpr int kBlock  = 512;            // 16 wave32s on one WGP
constexpr int kPPT    = kN / kBlock;    // 64 points per thread
constexpr int kChunks = kPPT / 4;       // 16 chunks of 4 points (3 x float4 each)
constexpr int kWaves  = kBlock / 32;    // 16

// Chunks per thread staged into LDS once via async load-to-LDS and re-scanned
// from DS every step (2 chunks = 8 points = 96B/thread -> 48KB total).
constexpr int kLdsChunks = 2;
constexpr int kLdsFloatsPerThr = kLdsChunks * 12;              // 24 floats
constexpr int kLdsFloats = kBlock * kLdsFloatsPerThr;          // 12288 floats

// ---------------------------------------------------------------------------
// FPS: one workgroup per batch. dmin lives in VGPRs (fully unrolled).
// A 48KB slice of xyz is staged to LDS once with global_load_async_to_lds_b128
// (ASYNCcnt); the rest streams from L2 every step. Per-step reduction is a
// single-barrier, coordinate-carrying xor-butterfly all-reduce.
// ---------------------------------------------------------------------------
__global__ __launch_bounds__(kBlock) void fps_kernel(
    const float* __restrict__ xyz,      // (B, N, 3)
    float* __restrict__ out_xyz,        // (B, S, 3)
    float* __restrict__ out_idx) {      // (B, S) stored as float
  const int b    = blockIdx.x;
  const int t    = threadIdx.x;
  const int lane = t & 31;
  const int wave = t >> 5;

  const float* px = xyz + (size_t)b * (size_t)kN * 3;

  __shared__ __align__(16) float sPts[kLdsFloats];   // staged xyz slice
  // Double-buffered wave-winner slots (single barrier per step, race-free).
  __shared__ float sV[2][kWaves];
  __shared__ int   sI[2][kWaves];
  __shared__ float sX[2][kWaves];
  __shared__ float sY[2][kWaves];
  __shared__ float sZ[2][kWaves];

  // Running min squared-distance buffer, register resident.
  float dmin[kPPT];
#pragma unroll
  for (int i = 0; i < kPPT; ++i) dmin[i] = 1e10f;

  const int p0 = t * kPPT;                       // first point owned by thread
  const float4* cb = (const float4*)(px + (size_t)p0 * 3);  // 16B aligned tile

  // gfx1250 prefetch (global_prefetch_b8): warm this thread's 768B tile.
  __builtin_prefetch((const void*)cb, 0, 3);
  __builtin_prefetch((const void*)((const char*)cb + 384), 0, 3);

  // ---- Async-stage the first kLdsChunks chunks of this thread's tile into
  // LDS (gfx1250 GLOBAL_LOAD_ASYNC_TO_LDS_B128, tracked by ASYNCcnt). Each
  // thread stages exactly the region it will read back. -------------------
  {
    const uint32_t ldsAddr = (uint32_t)(uintptr_t)(&sPts[t * kLdsFloatsPerThr]);
    const uint32_t gOff    = (uint32_t)(p0 * 12);           // byte offset
    const uint64_t gBase   = (uint64_t)(uintptr_t)px;
#pragma unroll
    for (int k = 0; k < kLdsChunks * 3; ++k) {              // 6 x 16B
      asm volatile("global_load_async_to_lds_b128 %0, %1, %2"
                   :
                   : "v"(ldsAddr + 16u * k), "v"(gOff + 16u * k), "s"(gBase)
                   : "memory");
    }
    asm volatile("s_wait_asynccnt 0" ::: "memory");
  }
  __syncthreads();

  // First selected index is 0 (reference convention).
  float qx = px[0], qy = px[1], qz = px[2];
  if (t == 0) {
    out_idx[(size_t)b * kS] = 0.0f;
    float* ox = out_xyz + (size_t)b * kS * 3;
    ox[0] = qx; ox[1] = qy; ox[2] = qz;
  }

  const float4* lb = (const float4*)(&sPts[t * kLdsFloatsPerThr]);

  for (int s = 1; s < kS; ++s) {
    float bestV = -1.0f;
    int   bestI = 0;

    // 4 AoS points per chunk = 48B = 3 x b128 (ds_load_b128 / global_load_b128)
    auto scan4 = [&](float4 f0, float4 f1, float4 f2, int c) {
      const float X[4] = {f0.x, f0.w, f1.z, f2.y};
      const float Y[4] = {f0.y, f1.x, f1.w, f2.z};
      const float Z[4] = {f0.z, f1.y, f2.x, f2.w};
#pragma unroll
      for (int j = 0; j < 4; ++j) {
        const float dx = X[j] - qx;
        const float dy = Y[j] - qy;
        const float dz = Z[j] - qz;
        const float d  = fmaf(dz, dz, fmaf(dy, dy, dx * dx));
        const float dm = fminf(dmin[c * 4 + j], d);
        dmin[c * 4 + j] = dm;
        if (dm > bestV) {            // strict >: keeps first occurrence
          bestV = dm;
          bestI = p0 + c * 4 + j;
        }
      }
    };

#pragma unroll
    for (int c = 0; c < kLdsChunks; ++c)          // LDS-resident chunks
      scan4(lb[c * 3 + 0], lb[c * 3 + 1], lb[c * 3 + 2], c);
#pragma unroll
    for (int c = kLdsChunks; c < kChunks; ++c)    // L2-streamed chunks
      scan4(cb[c * 3 + 0], cb[c * 3 + 1], cb[c * 3 + 2], c);

    // Fetch this thread's candidate coords now (parallel phase; the shuffles
    // below partially hide the L2 latency). Carried through the reduction so
    // the serial tail never touches global memory.
    const float* cp = px + (size_t)bestI * 3;
    float bx = cp[0], by = cp[1], bz = cp[2];

    // Stage 1: wave32 xor-butterfly all-reduce over (v, i, x, y, z).
    // Ties: lower index wins (jnp.argmax first-occurrence semantics).
#pragma unroll
    for (int off = 16; off > 0; off >>= 1) {
      const float ov = __shfl_xor(bestV, off, 32);
      const int   oi = __shfl_xor(bestI, off, 32);
      const float ox = __shfl_xor(bx, off, 32);
      const float oy = __shfl_xor(by, off, 32);
      const float oz = __shfl_xor(bz, off, 32);
      if (ov > bestV || (ov == bestV && oi < bestI)) {
        bestV = ov; bestI = oi; bx = ox; by = oy; bz = oz;
      }
    }

    const int buf = s & 1;
    if (lane == 0) {
      sV[buf][wave] = bestV; sI[buf][wave] = bestI;
      sX[buf][wave] = bx;    sY[buf][wave] = by;   sZ[buf][wave] = bz;
    }
    __syncthreads();   // single barrier per step

    // Stage 2: every wave redundantly all-reduces the 16 wave winners.
    // Entries are duplicated across lane halves, so start at off=8.
    const int e = lane & (kWaves - 1);
    float v = sV[buf][e];
    int   i = sI[buf][e];
    float x = sX[buf][e], y = sY[buf][e], z = sZ[buf][e];
#pragma unroll
    for (int off = kWaves / 2; off > 0; off >>= 1) {
      const float ov = __shfl_xor(v, off, 32);
      const int   oi = __shfl_xor(i, off, 32);
      const float ox = __shfl_xor(x, off, 32);
      const float oy = __shfl_xor(y, off, 32);
      const float oz = __shfl_xor(z, off, 32);
      if (ov > v || (ov == v && oi < i)) {
        v = ov; i = oi; x = ox; y = oy; z = oz;
      }
    }

    // Every thread now holds the step winner: new query, no broadcast needed.
    qx = x; qy = y; qz = z;
    if (t == 0) {
      out_idx[(size_t)b * kS + s] = (float)i;
      float* ox2 = out_xyz + ((size_t)b * kS + s) * 3;
      ox2[0] = x; ox2[1] = y; ox2[2] = z;
    }
  }
}

// ---------------------------------------------------------------------------
// Feature gather: out_fea[b][c][s] = features[b][c][idx[b][s]]
// One thread per output element; features are L2-resident (96 MB < 192 MB).
// ---------------------------------------------------------------------------
__global__ __launch_bounds__(256) void gather_kernel(
    const float* __restrict__ fea,      // (B, C, N)
    const float* __restrict__ idxF,     // (B, S) as float
    float* __restrict__ out_fea) {      // (B, C, S)
  const int tid = blockIdx.x * blockDim.x + threadIdx.x;
  if (tid >= kB * kC * kS) return;
  const int s  = tid % kS;
  const int bc = tid / kS;
  const int c  = bc % kC;
  const int b  = bc / kC;
  const int idx = (int)idxF[(size_t)b * kS + s];
  out_fea[tid] = fea[((size_t)b * kC + c) * (size_t)kN + idx];
}

// ---------------------------------------------------------------------------
extern "C" void kernel_launch(void* const* d_in, const int* in_sizes, int n_in,
                              void* d_out, int out_size, void* d_ws,
                              size_t ws_size, hipStream_t stream) {
  (void)in_sizes; (void)n_in; (void)d_ws; (void)ws_size; (void)out_size;

  const float* points_xyz = (const float*)d_in[0];   // (8, 32768, 3) f32
  const float* features   = (const float*)d_in[1];   // (8, 96, 32768) f32

  float* out      = (float*)d_out;
  float* out_xyz  = out;                               // 8*1024*3   = 24576
  float* out_fea  = out + (size_t)kB * kS * 3;         // 8*96*1024  = 786432
  float* out_idx  = out_fea + (size_t)kB * kC * kS;    // 8*1024     = 8192

  // Sequential FPS: one WGP per batch.
  fps_kernel<<<kB, kBlock, 0, stream>>>(points_xyz, out_xyz, out_idx);

  // Gather features by the selected indices (reads out_idx written above).
  const int total = kB * kC * kS;
  gather_kernel<<<(total + 255) / 256, 256, 0, stream>>>(features, out_idx,
                                                         out_fea);
}